// CLIPVisionTower_EfficientUICoder_10471130268336
// MI455X (gfx1250) — compile-verified
//
#include <hip/hip_runtime.h>
#include <hip/hip_bf16.h>
#include <stdint.h>

#define B_ 32
#define H_ 16
#define T_ 577
#define D_ 1024
#define P_ 576

typedef __attribute__((ext_vector_type(2))) float v2f;
typedef __attribute__((ext_vector_type(8))) float v8f;

// ---------------------------------------------------------------------------
// Kernel 1: bulk copy of hidden_states through LDS using CDNA5 async ops.
// Each block moves exactly 1024 x 16B chunks (4 per thread). 4616 blocks cover
// 32*577*1024 floats exactly (75,628,544 B / 16 = 4,726,784 = 4616*1024).
// ---------------------------------------------------------------------------
__global__ __launch_bounds__(256) void hs_copy_async(const float* __restrict__ src,
                                                     float* __restrict__ dst) {
  __shared__ __align__(16) unsigned char smem[256 * 4 * 16]; // 16 KB staging
  const unsigned tid = threadIdx.x;
  const unsigned long long base = (unsigned long long)blockIdx.x * (1024ull * 16ull);
  const unsigned long long s0 = (unsigned long long)(uintptr_t)src + base;
  const unsigned long long d0 = (unsigned long long)(uintptr_t)dst + base;

#pragma unroll
  for (int k = 0; k < 4; ++k) {
    unsigned lds = (unsigned)(uintptr_t)(smem + (k * 256 + tid) * 16);
    unsigned long long ga = s0 + (unsigned long long)(k * 256 + tid) * 16ull;
    asm volatile("global_load_async_to_lds_b128 %0, %1, off"
                 :: "v"(lds), "v"(ga) : "memory");
  }
  asm volatile("s_wait_asynccnt 0x0" ::: "memory");   // loads landed in LDS
#pragma unroll
  for (int k = 0; k < 4; ++k) {
    unsigned lds = (unsigned)(uintptr_t)(smem + (k * 256 + tid) * 16);
    unsigned long long ga = d0 + (unsigned long long)(k * 256 + tid) * 16ull;
    asm volatile("global_store_async_from_lds_b128 %0, %1, off"
                 :: "v"(ga), "v"(lds) : "memory");
  }
  // s_endpgm performs an implicit wait-idle; stores complete before kernel end.
}

// ---------------------------------------------------------------------------
// Kernel 2: cls_scores[b,p] = sum over 16 heads of attn[b,h,0,1+p] via WMMA.
// One wave handles one (batch, 16-token tile). A = all-ones 16x4 f32, so
// D[m,n] = sum_k B[k,n]; 4 chained v_wmma_f32_16x16x4_f32 reduce 16 heads.
// Lane L (n = L%16, half = L/16) supplies heads {4*hg + 2*half + 0,1} per step,
// covering all 16 heads exactly once per column. Order within a column is
// irrelevant for a sum, so only the N = lane%16 mapping matters.
// ---------------------------------------------------------------------------
__global__ __launch_bounds__(256) void cls_scores_wmma(const float* __restrict__ attn,
                                                       float* __restrict__ scores) {
  const unsigned wave = blockIdx.x * 8u + (threadIdx.x >> 5);  // 0..1151
  const unsigned lane = threadIdx.x & 31u;
  const unsigned b    = wave / 36u;
  const unsigned tile = wave % 36u;
  const unsigned n    = lane & 15u;
  const unsigned half = lane >> 4;
  const unsigned tok  = 1u + tile * 16u + n;   // column in attn row q==0
  const unsigned TT   = (unsigned)T_ * (unsigned)T_;

  const v2f ones = {1.0f, 1.0f};
  v8f c = {};
#pragma unroll
  for (int hg = 0; hg < 4; ++hg) {
    unsigned h0 = (unsigned)hg * 4u + half * 2u;
    const float* p = attn + ((unsigned long long)(b * H_ + h0) * TT + tok);
    v2f bm;
    bm.x = p[0];        // head h0
    bm.y = p[TT];       // head h0+1
    c = __builtin_amdgcn_wmma_f32_16x16x4_f32(false, ones, false, bm,
                                              (short)0, c, false, false);
  }
  // Every row of D is the column sum; row M=0 lives in c[0] of lanes 0..15.
  if (lane < 16u) scores[b * P_ + tile * 16u + n] = c[0];
}

// ---------------------------------------------------------------------------
// Kernel 3: per-batch token selection. One block per batch, one thread/patch.
// Stable ranks reproduce jnp.argsort tie-breaking (index ascending).
// ---------------------------------------------------------------------------
__global__ __launch_bounds__(576) void select_tokens(const float* __restrict__ scores,
                                                     const int* __restrict__ labels,
                                                     const int* __restrict__ tr_ptr,
                                                     float* __restrict__ mask_out) {
  __shared__ float s[P_];
  __shared__ int   lab[P_];
  __shared__ int   cnt[3];   // [0]=bg(-1) [1]=text(0) [2]=comp(1)
  const int b = blockIdx.x;
  const int i = threadIdx.x;
  if (i < 3) cnt[i] = 0;
  __syncthreads();

  const float si = scores[b * P_ + i];
  const int   li = labels[b * P_ + i];
  s[i] = si;
  lab[i] = li;
  atomicAdd(&cnt[li + 1], 1);
  __syncthreads();

  const int n_bg = cnt[0], n_text = cnt[1], n_comp = cnt[2];
  const int tr = tr_ptr[0];
  const int a  = min(min(tr, n_comp + n_text), n_bg);
  const int k1 = min(a, n_comp);
  const int k2 = min(a - k1, n_text);
  const int kb = min(k1 + k2, n_bg);

  // stable rank within own class (ascending for comp/text, descending for bg)
  const bool is_bg = (li == -1);
  int rank = 0;
  for (int j = 0; j < P_; ++j) {
    if (lab[j] == li) {
      const float sj = s[j];
      bool better;
      if (is_bg) better = (sj > si) || (sj == si && j < i);
      else       better = (sj < si) || (sj == si && j < i);
      rank += better ? 1 : 0;
    }
  }

  bool keep;
  if (li == 1)      keep = (rank >= k1);  // comp: drop k1 lowest
  else if (li == 0) keep = (rank >= k2);  // text: drop k2 lowest
  else              keep = (rank < kb);   // bg:   add kb highest
  mask_out[b * T_ + 1 + i] = keep ? 1.0f : 0.0f;
  if (i == 0) mask_out[b * T_] = 1.0f;    // CLS always kept
}

// ---------------------------------------------------------------------------
extern "C" void kernel_launch(void* const* d_in, const int* in_sizes, int n_in,
                              void* d_out, int out_size, void* d_ws, size_t ws_size,
                              hipStream_t stream) {
  const float* hs     = (const float*)d_in[0];   // [B,T,D] f32
  const float* attn   = (const float*)d_in[1];   // [B,H,T,T] f32
  const int*   labels = (const int*)d_in[2];     // [B,P] i32
  const int*   tr     = (const int*)d_in[3];     // scalar i32
  float* out    = (float*)d_out;                 // [B*T*D] hs copy, then [B*T] mask
  float* scores = (float*)d_ws;                  // B*P f32 scratch (73,728 B)

  (void)in_sizes; (void)n_in; (void)out_size; (void)ws_size;

  hs_copy_async<<<4616, 256, 0, stream>>>(hs, out);
  cls_scores_wmma<<<144, 256, 0, stream>>>(attn, scores);
  select_tokens<<<32, 576, 0, stream>>>(scores, labels, tr,
                                        out + (size_t)B_ * T_ * D_);
}